// ShapeletNet_65755949302130
// MI455X (gfx1250) — compile-verified
//
#include <hip/hip_runtime.h>
#include <hip/hip_bf16.h>
#include <math.h>

// Problem constants (from reference): B=64, W=256, N=128, L=64
#define SB 64
#define SW 256
#define SN 128
#define SL 64
#define SM (SB * SW)      // 16384 rows of x
#define TILE_M 16
#define TILE_N 16

typedef float v2f __attribute__((ext_vector_type(2)));
typedef float v8f __attribute__((ext_vector_type(8)));

// ---------------------------------------------------------------------------
// Kernel 1: initialize per-(b,n) min buffer to +inf (bit pattern 0x7F800000)
// ---------------------------------------------------------------------------
__global__ void shapelet_init_min(unsigned int* __restrict__ minbuf) {
    int idx = blockIdx.x * blockDim.x + threadIdx.x;
    if (idx < SB * SN) minbuf[idx] = 0x7F800000u;   // +inf
}

// ---------------------------------------------------------------------------
// Kernel 2: main distance kernel.
//   grid.x = SM/16 = 1024 row tiles, block = 256 threads = 8 waves.
//   x-tile (16x64 f32 = 4 KB) is staged into LDS with the CDNA5 async
//   global->LDS path (GLOBAL_LOAD_ASYNC_TO_LDS_B128, ASYNCcnt), overlapped
//   with each wave's shapelet register preload. Wave w then computes the
//   16x16 tile (rows r0..r0+15) x (cols 16w..16w+15) of dot = X * S^T via
//   16 chained V_WMMA_F32_16X16X4_F32 (K=64), reconstructs
//   discrim = sqrt(xx - 2*dot + ss), min-reduces over the 16 rows, and
//   atomicMin's into minbuf[b][n].
// ---------------------------------------------------------------------------
__global__ __launch_bounds__(256)
void shapelet_min_dist(const float* __restrict__ x,
                       const float* __restrict__ shapelets,
                       unsigned int* __restrict__ minbuf) {
    __shared__ float lds_x[TILE_M * SL];   // 16 rows x 64 floats = 4 KB
    __shared__ float lds_xx[TILE_M];       // ||x_row||^2 per tile row

    const int tid  = threadIdx.x;
    const int lane = tid & 31;             // wave32
    const int wave = tid >> 5;             // 0..7 -> column tile
    const int row_tile = blockIdx.x;       // 0..1023
    const int r0 = row_tile * TILE_M;      // first global row
    const int b_idx = r0 / SW;             // batch index (tiles never cross b)

    // Warm the cache for the (tiny, heavily reused) shapelet table.
    __builtin_prefetch(shapelets + (tid << 5), 0, 1);   // global_prefetch_b8

    // ---- Async-stage the 16x64 x-tile into LDS (16 B per lane) ----------
    // Generic shared-aperture address: low 32 bits are the LDS byte offset.
    {
        unsigned lds_dst = (unsigned)(uintptr_t)(&lds_x[0]) + (unsigned)(tid << 4);
        const float* gsrc = x + (size_t)r0 * SL + (tid << 2);
        asm volatile("global_load_async_to_lds_b128 %0, %1, off"
                     :
                     : "v"(lds_dst), "v"(gsrc)
                     : "memory");
    }

    // ---- Per-lane WMMA operand geometry (ISA 7.12.2, 32-bit A 16x4) ----
    // A: VGPR0 lanes 0-15 -> K=k0, lanes 16-31 -> K=k0+2 ; VGPR1 -> +1
    // B (4x16): same pattern with row = n.
    const int half = lane >> 4;            // 0: lanes 0-15, 1: lanes 16-31
    const int mrow = lane & 15;            // A-matrix row within tile
    const int ncol = (wave << 4) + (lane & 15);   // global shapelet index n

    // Preload this lane's half of shapelet row n: 16 x float2
    // (overlaps with the in-flight async LDS copy above)
    v2f sb[16];
    #pragma unroll
    for (int k = 0; k < 16; ++k) {
        const int base = (k << 2) + (half << 1);      // 4k + 2*half
        sb[k] = *reinterpret_cast<const v2f*>(shapelets + ncol * SL + base);
    }
    // Partial ||s_n||^2 over this lane's 32 elements; other half via xor-16
    float ssp = 0.f;
    #pragma unroll
    for (int k = 0; k < 16; ++k) ssp += sb[k].x * sb[k].x + sb[k].y * sb[k].y;
    const float ss_n = ssp + __shfl_xor(ssp, 16, 32);

    // Drain the async copy (ASYNCcnt is NOT covered by __syncthreads' waits)
    asm volatile("s_wait_asynccnt 0x0" ::: "memory");
    __syncthreads();

    // ---- Row norms ||x_row||^2 (threads 0..15, one row each; pk_fma'd) ----
    if (tid < TILE_M) {
        float acc = 0.f;
        const float* r = &lds_x[tid * SL];
        #pragma unroll
        for (int l = 0; l < SL; ++l) acc += r[l] * r[l];
        lds_xx[tid] = acc;
    }
    __syncthreads();

    // ---- GEMM accumulation: 16 x V_WMMA_F32_16X16X4_F32 ----
    v8f c = {};
    #pragma unroll
    for (int k = 0; k < 16; ++k) {
        const int base = (k << 2) + (half << 1);
        v2f a = *reinterpret_cast<const v2f*>(&lds_x[mrow * SL + base]);
        c = __builtin_amdgcn_wmma_f32_16x16x4_f32(
                /*neg_a=*/false, a, /*neg_b=*/false, sb[k],
                /*c_mod=*/(short)0, c, /*reuse_a=*/false, /*reuse_b=*/false);
    }

    // ---- discrim + min over the 16 rows of this tile ----
    // D layout: VGPR i, lanes 0-15 -> M=i, lanes 16-31 -> M=i+8, N=lane&15.
    float lmin = __builtin_inff();
    #pragma unroll
    for (int i = 0; i < 8; ++i) {
        const int m = i + (half << 3);
        float arg = lds_xx[m] - 2.0f * c[i] + ss_n;
        arg = fmaxf(arg, 0.0f);            // guard tiny negative rounding
        lmin = fminf(lmin, sqrtf(arg));
    }
    lmin = fminf(lmin, __shfl_xor(lmin, 16, 32));   // combine M 0-7 with 8-15

    // Non-negative floats: uint ordering == float ordering.
    if (half == 0) {
        atomicMin(&minbuf[b_idx * SN + ncol], __float_as_uint(lmin));
    }
}

// ---------------------------------------------------------------------------
// Kernel 3: logits[b] = min_discrim[b,:] . cls_w + cls_b ; sigmoid.
//   64 blocks x 128 threads (4 waves), shuffle + LDS reduction.
// ---------------------------------------------------------------------------
__global__ __launch_bounds__(128)
void shapelet_classify(const unsigned int* __restrict__ minbuf,
                       const float* __restrict__ cls_w,
                       const float* __restrict__ cls_b,
                       float* __restrict__ out) {
    __shared__ float lds_r[4];
    const int b = blockIdx.x;
    const int n = threadIdx.x;             // 0..127
    const int lane = n & 31;
    const int wave = n >> 5;

    float p = __uint_as_float(minbuf[b * SN + n]) * cls_w[n];
    #pragma unroll
    for (int off = 16; off > 0; off >>= 1) p += __shfl_down(p, off, 32);
    if (lane == 0) lds_r[wave] = p;
    __syncthreads();
    if (n == 0) {
        float logit = lds_r[0] + lds_r[1] + lds_r[2] + lds_r[3] + cls_b[0];
        out[b] = 1.0f / (1.0f + expf(-logit));
    }
}

// ---------------------------------------------------------------------------
extern "C" void kernel_launch(void* const* d_in, const int* in_sizes, int n_in,
                              void* d_out, int out_size, void* d_ws, size_t ws_size,
                              hipStream_t stream) {
    const float* x         = (const float*)d_in[0];   // [B, W, L]
    const float* shapelets = (const float*)d_in[1];   // [N, L]
    const float* cls_w     = (const float*)d_in[2];   // [1, N]
    const float* cls_b     = (const float*)d_in[3];   // [1]
    float* out = (float*)d_out;                       // [B, 1]

    unsigned int* minbuf = (unsigned int*)d_ws;       // B*N = 8192 u32 (32 KB)

    shapelet_init_min<<<(SB * SN + 255) / 256, 256, 0, stream>>>(minbuf);
    shapelet_min_dist<<<SM / TILE_M, 256, 0, stream>>>(x, shapelets, minbuf);
    shapelet_classify<<<SB, 128, 0, stream>>>(minbuf, cls_w, cls_b, out);
}